// PerceptionTransformer_28467043238173
// MI455X (gfx1250) — compile-verified
//
#include <hip/hip_runtime.h>
#include <hip/hip_bf16.h>
#include <math.h>

// ---------------- problem constants ----------------
#define LQ      22500
#define CDIM    256
#define NHEADS  8
#define DHEAD   32      // CDIM / NHEADS
#define NPOINTS 8
#define NCAMS   6
#define GH      58
#define GW      100
#define SLEN    (GH*GW)           // 5800
#define VROWS   (NCAMS*SLEN)      // 34800

typedef __attribute__((ext_vector_type(2))) float v2f;
typedef __attribute__((ext_vector_type(8))) float v8f;

// =====================================================================
// Generic f32 GEMM:  Out[M,N] = X[M,K=256] * W[N,K=256]^T + bias[N]
// via V_WMMA_F32_16X16X4_F32. One wave per 16x16 tile, 8 waves/block.
//
// f32 16x16x4 fragment layout (ISA 7.12.2):
//   A (16x4):  lane L: M = L&15, VGPR v holds K = v + 2*(L>>4)
//   B (4x16):  lane L: N = L&15, VGPR v holds K = v + 2*(L>>4)   (B[k][n]=W[n][k])
//   C/D (16x16): lane L: N = L&15, VGPR r holds M = r + 8*(L>>4)
//
// Fast path for full tiles (the overwhelming majority): no A-masking
// VALU in the K-loop, unguarded stores. Edge path only for the single
// partial M-tile when M % 16 != 0.
// =====================================================================
__global__ __launch_bounds__(256) void gemm_wmma_f32(
    const float* __restrict__ X, const float* __restrict__ W,
    const float* __restrict__ bias, float* __restrict__ Out,
    int M, int N, int ntiles, int tiles)
{
  const int K = CDIM;
  const int wave = threadIdx.x >> 5;
  const int lane = threadIdx.x & 31;
  const int t = blockIdx.x * 8 + wave;
  if (t >= tiles) return;                 // uniform per wave: EXEC stays all-1s

  const int mt = t / ntiles;
  const int nt = t - mt * ntiles;
  const int m0 = mt << 4, n0 = nt << 4;
  const int lo = lane & 15;
  const int hi = lane >> 4;

  const float* __restrict__ brow = W + (size_t)(n0 + lo) * K + 2 * hi;

  const float bv = bias[n0 + lo];
  v8f c;
#pragma unroll
  for (int r = 0; r < 8; ++r) c[r] = bv;

  if (m0 + 16 <= M) {
    // ---------------- full tile (hot path) ----------------
    const float* __restrict__ arow = X + (size_t)(m0 + lo) * K + 2 * hi;
#pragma unroll 4
    for (int k = 0; k < K; k += 4) {
      const v2f a = *(const v2f*)(arow + k);   // 8B aligned
      const v2f b = *(const v2f*)(brow + k);
      // 8 args: (neg_a, A, neg_b, B, c_mod, C, reuse_a, reuse_b)
      c = __builtin_amdgcn_wmma_f32_16x16x4_f32(false, a, false, b, (short)0, c,
                                                false, false);
    }
    float* __restrict__ orow = Out + (size_t)(m0 + 8 * hi) * N + n0 + lo;
#pragma unroll
    for (int r = 0; r < 8; ++r) orow[(size_t)r * N] = c[r];
  } else {
    // ---------------- edge tile (M % 16 rows valid) ----------------
    const int   m     = m0 + lo;
    const float amask = (m < M) ? 1.0f : 0.0f;
    const int   mrd   = (m < M) ? m : (M - 1);
    const float* __restrict__ arow = X + (size_t)mrd * K + 2 * hi;
#pragma unroll 4
    for (int k = 0; k < K; k += 4) {
      v2f a = *(const v2f*)(arow + k);
      const v2f b = *(const v2f*)(brow + k);
      a.x *= amask; a.y *= amask;
      c = __builtin_amdgcn_wmma_f32_16x16x4_f32(false, a, false, b, (short)0, c,
                                                false, false);
    }
#pragma unroll
    for (int r = 0; r < 8; ++r) {
      const int mm = m0 + r + 8 * hi;
      if (mm < M) Out[(size_t)mm * N + n0 + lo] = c[r];
    }
  }
}

// =====================================================================
// Fused deformable sampling: softmax(logits), offset add, bilinear
// gather over 6 cams * 8 points * 4 taps, vis-mask + count normalize.
// Block = 1 query (256 thr = 8 waves); wave = head; lane = channel dim.
// val is (cam, y, x, head, dim) so taps are 128B-coalesced per wave.
// =====================================================================
__global__ __launch_bounds__(256) void deform_sample_kernel(
    const float* __restrict__ val,     // (6, 5800, 256)
    const float* __restrict__ offs,    // (22500, 128) = (q, h, p, 2)
    const float* __restrict__ logits,  // (22500, 64)  = (q, h, p)
    const float* __restrict__ ref,     // (6, 1, 22500, 4, 2)
    const int*   __restrict__ mask,    // (6, 1, 22500, 4)
    float* __restrict__ slots)         // (22500, 256)
{
  const int q    = blockIdx.x;
  const int h    = threadIdx.x >> 5;
  const int lane = threadIdx.x & 31;

  // ---- softmax over the 8 points of this (q, h) ----
  const float* lg = logits + (size_t)q * (NHEADS * NPOINTS) + h * NPOINTS;
  float w8[NPOINTS];
  float mx = -3.402823466e+38f;
#pragma unroll
  for (int p = 0; p < NPOINTS; ++p) { w8[p] = lg[p]; mx = fmaxf(mx, w8[p]); }
  float sum = 0.0f;
#pragma unroll
  for (int p = 0; p < NPOINTS; ++p) { w8[p] = __expf(w8[p] - mx); sum += w8[p]; }
  const float rsum = 1.0f / sum;
#pragma unroll
  for (int p = 0; p < NPOINTS; ++p) w8[p] *= rsum;

  // ---- normalized offsets for this (q, h) ----
  const float* of = offs + (size_t)q * (NHEADS * NPOINTS * 2) + h * (NPOINTS * 2);
  float ox[NPOINTS], oy[NPOINTS];
#pragma unroll
  for (int p = 0; p < NPOINTS; ++p) {
    ox[p] = of[p * 2 + 0] * (1.0f / (float)GW);
    oy[p] = of[p * 2 + 1] * (1.0f / (float)GH);
  }

  float acc = 0.0f;
  float count = 0.0f;
  for (int cam = 0; cam < NCAMS; ++cam) {
    const int mbase = (cam * LQ + q) * 4;
    const int vsum = mask[mbase] + mask[mbase + 1] + mask[mbase + 2] + mask[mbase + 3];
    if (vsum <= 0) continue;             // uniform across the whole block
    count += 1.0f;

    const float* rp = ref + (size_t)(cam * LQ + q) * 8;          // 4 anchors * 2
    const float* vb = val + (size_t)cam * SLEN * CDIM + h * DHEAD + lane;

    float a = 0.0f;
#pragma unroll
    for (int p = 0; p < NPOINTS; ++p) {
      const int z = p & 3;               // Z-anchor index (p = g*4 + z)
      const float lx = rp[z * 2 + 0] + ox[p];
      const float ly = rp[z * 2 + 1] + oy[p];
      const float x = lx * (float)GW - 0.5f;
      const float y = ly * (float)GH - 0.5f;
      const float x0f = floorf(x), y0f = floorf(y);
      const float fx = x - x0f, fy = y - y0f;
      const int x0 = (int)x0f, y0 = (int)y0f;

      float s = 0.0f;
      {
        const int yi = y0, xi = x0;
        if (yi >= 0 && yi < GH && xi >= 0 && xi < GW)
          s += vb[(size_t)(yi * GW + xi) * CDIM] * ((1.0f - fy) * (1.0f - fx));
      }
      {
        const int yi = y0, xi = x0 + 1;
        if (yi >= 0 && yi < GH && xi >= 0 && xi < GW)
          s += vb[(size_t)(yi * GW + xi) * CDIM] * ((1.0f - fy) * fx);
      }
      {
        const int yi = y0 + 1, xi = x0;
        if (yi >= 0 && yi < GH && xi >= 0 && xi < GW)
          s += vb[(size_t)(yi * GW + xi) * CDIM] * (fy * (1.0f - fx));
      }
      {
        const int yi = y0 + 1, xi = x0 + 1;
        if (yi >= 0 && yi < GH && xi >= 0 && xi < GW)
          s += vb[(size_t)(yi * GW + xi) * CDIM] * (fy * fx);
      }
      a += w8[p] * s;
    }
    acc += a;
  }

  slots[(size_t)q * CDIM + h * DHEAD + lane] = acc / fmaxf(count, 1.0f);
}

// =====================================================================
static inline void launch_gemm(const float* X, const float* W, const float* bias,
                               float* Out, int M, int N, hipStream_t s) {
  const int mt = (M + 15) >> 4;
  const int nt = N >> 4;
  const int tiles = mt * nt;
  const int blocks = (tiles + 7) >> 3;   // 8 waves (tiles) per 256-thread block
  gemm_wmma_f32<<<blocks, 256, 0, s>>>(X, W, bias, Out, M, N, nt, tiles);
}

extern "C" void kernel_launch(void* const* d_in, const int* in_sizes, int n_in,
                              void* d_out, int out_size, void* d_ws, size_t ws_size,
                              hipStream_t stream) {
  const float* query  = (const float*)d_in[0];   // (1, 22500, 256)
  const float* ref    = (const float*)d_in[1];   // (6, 1, 22500, 4, 2)
  const int*   bmask  = (const int*)  d_in[2];   // (6, 1, 22500, 4)
  const float* value  = (const float*)d_in[3];   // (6, 5800, 256)
  const float* W_off  = (const float*)d_in[4];   // (128, 256)
  const float* b_off  = (const float*)d_in[5];   // (128,)
  const float* W_attn = (const float*)d_in[6];   // (64, 256)
  const float* b_attn = (const float*)d_in[7];   // (64,)
  const float* W_val  = (const float*)d_in[8];   // (256, 256)
  const float* b_val  = (const float*)d_in[9];   // (256,)
  const float* W_out  = (const float*)d_in[10];  // (256, 256)
  const float* b_out  = (const float*)d_in[11];  // (256,)
  float* out = (float*)d_out;                    // (1, 22500, 256)

  // workspace layout (f32): ~19.0M floats (~76 MB)
  float* val_proj = (float*)d_ws;                          // 34800*256
  float* offs     = val_proj + (size_t)VROWS * CDIM;       // 22500*128
  float* logits   = offs     + (size_t)LQ * 128;           // 22500*64
  float* slots    = logits   + (size_t)LQ * 64;            // 22500*256

  // 1) value projection  (cam*S, 256) x (256,256)^T
  launch_gemm(value, W_val, b_val, val_proj, VROWS, CDIM, stream);
  // 2) sampling-offset projection  (22500,256) x (256,128)^T
  launch_gemm(query, W_off, b_off, offs, LQ, 128, stream);
  // 3) attention-logit projection  (22500,256) x (256,64)^T
  launch_gemm(query, W_attn, b_attn, logits, LQ, 64, stream);
  // 4) fused softmax + deformable bilinear sampling + mask/count normalize
  deform_sample_kernel<<<LQ, 256, 0, stream>>>(val_proj, offs, logits, ref, bmask, slots);
  // 5) output projection  (22500,256) x (256,256)^T
  launch_gemm(slots, W_out, b_out, out, LQ, CDIM, stream);
}